// EquivariantGraphConv_58557584114045
// MI455X (gfx1250) — compile-verified
//
#include <hip/hip_runtime.h>
#include <math.h>

typedef __attribute__((ext_vector_type(2))) float v2f;
typedef __attribute__((ext_vector_type(8))) float v8f;

#define EPSV 1e-8f

__device__ __forceinline__ float sigmoidf_(float x) { return 1.0f / (1.0f + __expf(-x)); }

// One 16x16 output tile of (16 rows) @ B(64 x ldb picking 16 cols), K=64 via 16x WMMA f32 16x16x4.
// A[ks] holds the lane's slice of the 16x4 A fragment for k-step ks (per ISA layout).
__device__ __forceinline__ v8f wmma_tile_f32(const v2f A[16], const float* __restrict__ B,
                                             int ldb, int col, int kOff)
{
    v8f acc = (v8f)(0.0f);
#pragma unroll
    for (int ks = 0; ks < 16; ++ks) {
        int k = ks * 4 + kOff;  // kOff = 0 (lanes 0-15) or 2 (lanes 16-31)
        v2f b;
        b.x = B[(size_t)k * ldb + col];
        b.y = B[(size_t)(k + 1) * ldb + col];
        acc = __builtin_amdgcn_wmma_f32_16x16x4_f32(false, A[ks], false, b, (short)0, acc,
                                                    false, false);
    }
    return acc;
}

// D layout: VGPR v -> row v (lanes 0-15) / row v+8 (lanes 16-31); column = lane&15 (already in dst).
// Full tiles only: fully unconditional stores, no exec-mask churn.
__device__ __forceinline__ void store_tile(float* __restrict__ dst, int rowStride, int lane,
                                           v8f acc, float bias)
{
    int rbase = (lane < 16) ? 0 : 8;
#pragma unroll
    for (int v = 0; v < 8; ++v)
        dst[(size_t)(rbase + v) * rowStride] = acc[v] + bias;
}

__global__ void zero_f32_kernel(float* __restrict__ p, int n)
{
    int i = blockIdx.x * blockDim.x + threadIdx.x;
    if (i < n) p[i] = 0.0f;
}

// Wave-per-task node GEMMs over FULL 16-node tiles only (tail handled by node_tail_kernel).
// task = tile*4 + channel (0 = scalar row, 1..3 = vector components).
// scalar: S_r = s@W_sr (ws), out[ch0] = s@W_sroot + b_sroot
// vector: V_r = v@W_vr (ws), out[ch1+cc] = v@W_vroot, vec_geom = v@W_vg (ws)
__global__ void node_transform_kernel(
    const float* __restrict__ x,
    const float* __restrict__ W_sr, const float* __restrict__ W_sroot,
    const float* __restrict__ b_sroot,
    const float* __restrict__ W_vr, const float* __restrict__ W_vroot,
    const float* __restrict__ W_vg,
    float* __restrict__ S_r, float* __restrict__ V_r,
    float* __restrict__ vec_geom, float* __restrict__ out,
    int nTilesFull)
{
    int wave = (blockIdx.x * blockDim.x + threadIdx.x) >> 5;
    int lane = threadIdx.x & 31;
    int tile = wave >> 2;
    int ch   = wave & 3;
    if (tile >= nTilesFull) return;  // wave-uniform: EXEC stays all-ones for WMMA

    int node0 = tile * 16;
    int m     = lane & 15;
    int kOff  = (lane >> 4) << 1;    // 0 or 2

    // A fragment cache for all 16 k-steps: x row (node0+m, channel ch), 64 wide.
    const float* rowp = x + ((size_t)(node0 + m) * 4 + (size_t)ch) * 64;
    v2f A[16];
#pragma unroll
    for (int ks = 0; ks < 16; ++ks) {
        int k = ks * 4 + kOff;
        A[ks].x = rowp[k];
        A[ks].y = rowp[k + 1];
    }

    int col16 = lane & 15;
    if (ch == 0) {
        for (int nt = 0; nt < 4; ++nt) {
            int col = nt * 16 + col16;
            v8f a1 = wmma_tile_f32(A, W_sr, 64, col, kOff);
            store_tile(S_r + (size_t)node0 * 64 + col, 64, lane, a1, 0.0f);
            v8f a2 = wmma_tile_f32(A, W_sroot, 64, col, kOff);
            store_tile(out + (size_t)node0 * 256 + col, 256, lane, a2, b_sroot[col]);
        }
    } else {
        int cc = ch - 1;
        for (int nt = 0; nt < 4; ++nt) {
            int col = nt * 16 + col16;
            v8f a1 = wmma_tile_f32(A, W_vr, 64, col, kOff);
            store_tile(V_r + (size_t)node0 * 192 + (size_t)cc * 64 + col, 192, lane, a1, 0.0f);
            v8f a2 = wmma_tile_f32(A, W_vroot, 64, col, kOff);
            store_tile(out + (size_t)node0 * 256 + (size_t)(1 + cc) * 64 + col, 256, lane, a2,
                       0.0f);
        }
        v8f a3 = wmma_tile_f32(A, W_vg, 16, col16, kOff);
        store_tile(vec_geom + (size_t)node0 * 48 + (size_t)cc * 16 + col16, 48, lane, a3, 0.0f);
    }
}

// Scalar fallback for the last N%16 nodes (launched only when N is not a multiple of 16).
// One thread per (tail node, output column h in 0..63).
__global__ void node_tail_kernel(
    const float* __restrict__ x,
    const float* __restrict__ W_sr, const float* __restrict__ W_sroot,
    const float* __restrict__ b_sroot,
    const float* __restrict__ W_vr, const float* __restrict__ W_vroot,
    const float* __restrict__ W_vg,
    float* __restrict__ S_r, float* __restrict__ V_r,
    float* __restrict__ vec_geom, float* __restrict__ out,
    int n0, int N)
{
    int t = blockIdx.x * blockDim.x + threadIdx.x;
    int nTail = N - n0;
    if (t >= nTail * 64) return;
    int n = n0 + (t >> 6);
    int h = t & 63;

    const float* xs = x + (size_t)n * 256;
    float s_sr = 0.0f, s_rt = 0.0f;
    for (int k = 0; k < 64; ++k) {
        float s = xs[k];
        s_sr += s * W_sr[k * 64 + h];
        s_rt += s * W_sroot[k * 64 + h];
    }
    S_r[(size_t)n * 64 + h] = s_sr;
    out[(size_t)n * 256 + h] = s_rt + b_sroot[h];

    for (int cc = 0; cc < 3; ++cc) {
        const float* xv = xs + (1 + cc) * 64;
        float vr = 0.0f, vrt = 0.0f;
        for (int k = 0; k < 64; ++k) {
            float v = xv[k];
            vr  += v * W_vr[k * 64 + h];
            vrt += v * W_vroot[k * 64 + h];
        }
        V_r[(size_t)n * 192 + (size_t)cc * 64 + h] = vr;
        out[(size_t)n * 256 + (size_t)(1 + cc) * 64 + h] = vrt;
        if (h < 16) {
            float vg = 0.0f;
            for (int k = 0; k < 64; ++k) vg += xv[k] * W_vg[k * 16 + h];
            vec_geom[(size_t)n * 48 + (size_t)cc * 16 + h] = vg;
        }
    }
}

// Thread per (edge, g): e_ij direction and atomic scatter-add into u_node[col].
__global__ void edge_dir_scatter_kernel(const int* __restrict__ ei,
                                        const float* __restrict__ vec_geom,
                                        float* __restrict__ u_node, int E)
{
    int t = blockIdx.x * blockDim.x + threadIdx.x;
    if (t >= E * 16) return;
    int e = t >> 4, g = t & 15;
    int r = ei[e];
    int c = ei[E + e];
    const float* pr = vec_geom + (size_t)r * 48 + g;
    const float* pc = vec_geom + (size_t)c * 48 + g;
    float bx = pr[0] - pc[0];
    float by = pr[16] - pc[16];
    float bz = pr[32] - pc[32];
    float inv = 1.0f / fmaxf(sqrtf(bx * bx + by * by + bz * bz), EPSV);
    float* pu = u_node + (size_t)c * 48 + g;
    atomicAdd(pu,      bx * inv);
    atomicAdd(pu + 16, by * inv);
    atomicAdd(pu + 32, bz * inv);
}

// Wave per edge: geometry feats + MLPs (g = lane&15, both wave halves compute, EXEC full),
// shfl-broadcast m_s/m_v, each lane gates + scatters 2 of the 64 output channels.
__global__ void edge_message_kernel(
    const int* __restrict__ ei,
    const float* __restrict__ vec_geom, const float* __restrict__ u_node,
    const float* __restrict__ S_r, const float* __restrict__ V_r,
    const float* __restrict__ Ws1, const float* __restrict__ bs1,
    const float* __restrict__ Ws2, const float* __restrict__ bs2,
    const float* __restrict__ Wv1, const float* __restrict__ bv1,
    const float* __restrict__ Wv2, const float* __restrict__ bv2,
    const float* __restrict__ W_es, const float* __restrict__ b_es,
    const float* __restrict__ W_ev, const float* __restrict__ b_ev,
    float* __restrict__ out, int E)
{
    int e    = (blockIdx.x * blockDim.x + threadIdx.x) >> 5;
    int lane = threadIdx.x & 31;
    if (e >= E) return;

    int r = ei[e];
    int c = ei[E + e];
    int g = lane & 15;

    // recompute e_ij (cheaper than spilling E*48 floats to DRAM; vec_geom is L2-resident)
    const float* pr = vec_geom + (size_t)r * 48 + g;
    const float* pc = vec_geom + (size_t)c * 48 + g;
    float bx = pr[0] - pc[0], by = pr[16] - pc[16], bz = pr[32] - pc[32];
    float inv = 1.0f / fmaxf(sqrtf(bx * bx + by * by + bz * bz), EPSV);
    float ex = bx * inv, ey = by * inv, ez = bz * inv;

    const float* pui = u_node + (size_t)c * 48 + g;
    const float* puj = u_node + (size_t)r * 48 + g;
    float uix = pui[0], uiy = pui[16], uiz = pui[32];
    float ujx = puj[0], ujy = puj[16], ujz = puj[32];

    float dot_ui_e = uix * ex + uiy * ey + uiz * ez;
    float norm_ui  = fmaxf(sqrtf(uix * uix + uiy * uiy + uiz * uiz), EPSV);
    float ang      = fminf(fmaxf(dot_ui_e / norm_ui, -1.0f), 1.0f);

    float pix = uix - dot_ui_e * ex, piy = uiy - dot_ui_e * ey, piz = uiz - dot_ui_e * ez;
    float dot_uj_e = ujx * ex + ujy * ey + ujz * ez;
    float pjx = ujx - dot_uj_e * ex, pjy = ujy - dot_uj_e * ey, pjz = ujz - dot_uj_e * ez;
    float dotp = pix * pjx + piy * pjy + piz * pjz;
    float ni = fmaxf(sqrtf(pix * pix + piy * piy + piz * piz), EPSV);
    float nj = fmaxf(sqrtf(pjx * pjx + pjy * pjy + pjz * pjz), EPSV);
    float dih = fminf(fmaxf(dotp / fmaxf(ni * nj, EPSV), -1.0f), 1.0f);

    // tiny edge MLPs: (ang, dih) -> 16 -> 1 (silu hidden), weights wave-uniform (scalar cached)
    float ms = bs2[0];
    float mv = bv2[0];
#pragma unroll
    for (int j = 0; j < 16; ++j) {
        float hs = ang * Ws1[j] + dih * Ws1[16 + j] + bs1[j];
        ms += (hs * sigmoidf_(hs)) * Ws2[j];
        float hv = ang * Wv1[j] + dih * Wv1[16 + j] + bv1[j];
        mv += (hv * sigmoidf_(hv)) * Wv2[j];
    }

    // gate projection G=16 -> H=64: each lane handles columns h0=lane, h1=lane+32
    float gs0 = b_es[lane], gs1 = b_es[lane + 32];
    float gv0 = b_ev[lane], gv1 = b_ev[lane + 32];
#pragma unroll
    for (int gg = 0; gg < 16; ++gg) {
        float msg = __shfl(ms, gg, 32);
        float mvg = __shfl(mv, gg, 32);
        gs0 += msg * W_es[gg * 64 + lane];
        gs1 += msg * W_es[gg * 64 + lane + 32];
        gv0 += mvg * W_ev[gg * 64 + lane];
        gv1 += mvg * W_ev[gg * 64 + lane + 32];
    }
    gs0 = sigmoidf_(gs0); gs1 = sigmoidf_(gs1);
    gv0 = sigmoidf_(gv0); gv1 = sigmoidf_(gv1);

    const float* sr = S_r + (size_t)c * 64;
    const float* vr = V_r + (size_t)c * 192;
    float* po = out + (size_t)r * 256;

    atomicAdd(po + lane,      gs0 * sr[lane]);
    atomicAdd(po + lane + 32, gs1 * sr[lane + 32]);
#pragma unroll
    for (int cc = 0; cc < 3; ++cc) {
        atomicAdd(po + 64 + cc * 64 + lane,      gv0 * vr[cc * 64 + lane]);
        atomicAdd(po + 64 + cc * 64 + lane + 32, gv1 * vr[cc * 64 + lane + 32]);
    }
}

extern "C" void kernel_launch(void* const* d_in, const int* in_sizes, int n_in,
                              void* d_out, int out_size, void* d_ws, size_t ws_size,
                              hipStream_t stream)
{
    const float* x       = (const float*)d_in[0];
    const int*   ei      = (const int*)d_in[1];
    const float* W_sr    = (const float*)d_in[2];
    const float* W_sroot = (const float*)d_in[3];
    const float* b_sroot = (const float*)d_in[4];
    const float* W_vr    = (const float*)d_in[5];
    const float* W_vroot = (const float*)d_in[6];
    const float* W_vg    = (const float*)d_in[7];
    const float* Ws1     = (const float*)d_in[8];
    const float* bs1     = (const float*)d_in[9];
    const float* Ws2     = (const float*)d_in[10];
    const float* bs2     = (const float*)d_in[11];
    const float* Wv1     = (const float*)d_in[12];
    const float* bv1     = (const float*)d_in[13];
    const float* Wv2     = (const float*)d_in[14];
    const float* bv2     = (const float*)d_in[15];
    const float* W_es    = (const float*)d_in[16];
    const float* b_es    = (const float*)d_in[17];
    const float* W_ev    = (const float*)d_in[18];
    const float* b_ev    = (const float*)d_in[19];

    const int N = in_sizes[0] / 256;  // (N,4,64)
    const int E = in_sizes[1] / 2;    // (2,E)

    // Workspace layout (floats): vec_geom N*48 | u_node N*48 | S_r N*64 | V_r N*192  (~70 MB)
    float* ws       = (float*)d_ws;
    float* vec_geom = ws;
    float* u_node   = vec_geom + (size_t)N * 48;
    float* S_r      = u_node + (size_t)N * 48;
    float* V_r      = S_r + (size_t)N * 64;
    float* out      = (float*)d_out;

    // 1) zero the u_node accumulator (must be re-zeroed every call)
    int uElems = N * 48;
    zero_f32_kernel<<<(uElems + 255) / 256, 256, 0, stream>>>(u_node, uElems);

    // 2) node-level WMMA GEMMs over full tiles (also writes root/base terms into d_out)
    int nTilesFull = N / 16;
    if (nTilesFull > 0) {
        int threads1 = nTilesFull * 4 * 32;  // one wave per (tile, channel)
        node_transform_kernel<<<(threads1 + 255) / 256, 256, 0, stream>>>(
            x, W_sr, W_sroot, b_sroot, W_vr, W_vroot, W_vg, S_r, V_r, vec_geom, out,
            nTilesFull);
    }
    // 2b) scalar tail for N % 16 leftover nodes (never taken for N = 50000)
    int n0 = nTilesFull * 16;
    if (n0 < N) {
        int tTail = (N - n0) * 64;
        node_tail_kernel<<<(tTail + 255) / 256, 256, 0, stream>>>(
            x, W_sr, W_sroot, b_sroot, W_vr, W_vroot, W_vg, S_r, V_r, vec_geom, out, n0, N);
    }

    // 3) edge directions -> atomic scatter into u_node
    int t2 = E * 16;
    edge_dir_scatter_kernel<<<(t2 + 255) / 256, 256, 0, stream>>>(ei, vec_geom, u_node, E);

    // 4) edge features, gates, gated message scatter into d_out (one wave per edge)
    edge_message_kernel<<<(E + 7) / 8, 256, 0, stream>>>(
        ei, vec_geom, u_node, S_r, V_r, Ws1, bs1, Ws2, bs2, Wv1, bv1, Wv2, bv2,
        W_es, b_es, W_ev, b_ev, out, E);
}